// PhiMoE_38637525794984
// MI455X (gfx1250) — compile-verified
//
#include <hip/hip_runtime.h>
#include <hip/hip_bf16.h>

// Problem constants (from reference)
#define B_ 32768
#define D_ 784
#define H_ 85
#define O_ 10
#define E_ 24
#define KSEL 2

// Tiling: D = 14 * 56, K-chunk of 56 = 14 wmma k-steps of 4
#define KC 56
#define NCHUNK 14

// LDS strides chosen for conflict-free banking (64 banks x 4B):
//  A-tile stride 57:  57 == -7 mod 64 -> 16 distinct banks over rows, and the
//                     hi-half k+2 shift never collides with the {7d} diff set.
//  B-tile strides satisfy 2*S mod 64 == 16 so the two half-wave 16-bank
//                     windows (rows k and k+2) are disjoint.
//  H-tile stride 98:  34*r mod 64 distinct for r=0..15; k+2 half disjoint except hi row 15 vs lo row 0.
#define ASTRIDE 57
#define BSTRIDE 104   // w1 chunk stride (logical 96 cols)
#define HSTRIDE 98    // hidden tile stride (logical 96 cols)
#define GSTRIDE 40    // gate_w chunk stride (logical 32 cols)

typedef __attribute__((ext_vector_type(2))) float v2f;
typedef __attribute__((ext_vector_type(8))) float v8f;

// ---------------------------------------------------------------------------
// init: zero per-expert counters
// ---------------------------------------------------------------------------
__global__ void moe_init_kernel(int* __restrict__ counts) {
    if (threadIdx.x < E_) counts[threadIdx.x] = 0;
}

// ---------------------------------------------------------------------------
// router: gate GEMM (WMMA f32 16x16x4) + softmax + top-2 + scatter to lists
// block = 128 threads (4 waves), 64 tokens per block, grid = B/64
// ---------------------------------------------------------------------------
__global__ __launch_bounds__(128) void moe_router_kernel(
    const float* __restrict__ x,        // [B, D]
    const float* __restrict__ gw,       // [D, E]
    const float* __restrict__ gb,       // [E]
    float* __restrict__ gate_probs,     // [B, E] (output region 2)
    int* __restrict__ counts,           // [E]
    int* __restrict__ lists,            // [E, B]
    int* __restrict__ pos,              // [B, 2]  -> e*B + slot
    float* __restrict__ topw)           // [B, 2]  -> normalized top-k weight
{
    __shared__ float Xs[64 * ASTRIDE];  // x tile
    __shared__ float Gs[KC * GSTRIDE];  // gate_w chunk, E padded to 32 (zeros)
    __shared__ float Ls[64 * 32];       // logits

    const int tid  = threadIdx.x;
    const int lane = tid & 31;
    const int wave = tid >> 5;
    const int lrow = lane & 15;     // row/col within 16
    const int lhi  = lane >> 4;     // k half-select
    const int tokBase = blockIdx.x * 64;

    v8f acc0 = {}, acc1 = {};

    for (int kc = 0; kc < NCHUNK; ++kc) {
        const int k0 = kc * KC;
        for (int i = tid; i < 64 * KC; i += 128) {
            int r = i / KC, c = i % KC;
            Xs[r * ASTRIDE + c] = x[(size_t)(tokBase + r) * D_ + k0 + c];
        }
        for (int i = tid; i < KC * 32; i += 128) {
            int kk = i >> 5, c = i & 31;
            Gs[kk * GSTRIDE + c] = (c < E_) ? gw[(size_t)(k0 + kk) * E_ + c]
                                            : 0.0f;
        }
        __syncthreads();
#pragma unroll
        for (int p = 0; p < 2; ++p) {
            const int pair = wave * 2 + p;     // 8 pairs: mt 0..3, nt 0..1
            const int mt = pair & 3, nt = pair >> 2;
            v8f* ap = p ? &acc1 : &acc0;
            const float* ar = &Xs[(mt * 16 + lrow) * ASTRIDE + lhi * 2];
            const float* bc = &Gs[(lhi * 2) * GSTRIDE + nt * 16 + lrow];
#pragma unroll
            for (int ks = 0; ks < NCHUNK; ++ks) {
                v2f a, b;
                a.x = ar[ks * 4 + 0];
                a.y = ar[ks * 4 + 1];
                b.x = bc[(ks * 4 + 0) * GSTRIDE];
                b.y = bc[(ks * 4 + 1) * GSTRIDE];
                *ap = __builtin_amdgcn_wmma_f32_16x16x4_f32(
                        false, a, false, b, (short)0, *ap, false, false);
            }
        }
        __syncthreads();
    }

    // spill logits (+bias) to LDS.  C layout: VGPR i -> M = i (+8 for hi lanes)
#pragma unroll
    for (int p = 0; p < 2; ++p) {
        const int pair = wave * 2 + p;
        const int mt = pair & 3, nt = pair >> 2;
        v8f acc = p ? acc1 : acc0;
        const int col = nt * 16 + lrow;
        const float bias = (col < E_) ? gb[col] : 0.0f;
#pragma unroll
        for (int i = 0; i < 8; ++i) {
            int row = mt * 16 + i + lhi * 8;
            Ls[row * 32 + col] = acc[i] + bias;
        }
    }
    __syncthreads();

    // per-token softmax + top-2 + scatter (threads 0..63)
    if (tid < 64) {
        const int btok = tokBase + tid;
        float m = -3.4e38f;
#pragma unroll
        for (int ee = 0; ee < E_; ++ee) m = fmaxf(m, Ls[tid * 32 + ee]);
        float pv[E_];
        float sum = 0.0f;
#pragma unroll
        for (int ee = 0; ee < E_; ++ee) {
            float v = __expf(Ls[tid * 32 + ee] - m);
            pv[ee] = v; sum += v;
        }
        const float inv = 1.0f / sum;
        float best0 = -1.0f, best1 = -1.0f;
        int i0 = 0, i1 = 0;
#pragma unroll
        for (int ee = 0; ee < E_; ++ee) {
            gate_probs[(size_t)btok * E_ + ee] = pv[ee] * inv;
            if (pv[ee] > best0) { best1 = best0; i1 = i0; best0 = pv[ee]; i0 = ee; }
            else if (pv[ee] > best1) { best1 = pv[ee]; i1 = ee; }
        }
        const float wnorm = 1.0f / (best0 + best1);
        int s0 = atomicAdd(&counts[i0], 1);
        lists[i0 * B_ + s0] = btok;
        pos[btok * 2 + 0]  = i0 * B_ + s0;
        topw[btok * 2 + 0] = best0 * wnorm;
        int s1 = atomicAdd(&counts[i1], 1);
        lists[i1 * B_ + s1] = btok;
        pos[btok * 2 + 1]  = i1 * B_ + s1;
        topw[btok * 2 + 1] = best1 * wnorm;
    }
}

// ---------------------------------------------------------------------------
// expert: per-expert gathered two-stage MLP, fully WMMA f32
// grid = (E, B/64), block = 256 (8 waves), 64 token-slots per block
// ---------------------------------------------------------------------------
__global__ __launch_bounds__(256) void moe_expert_kernel(
    const float* __restrict__ x,        // [B, D]
    const float* __restrict__ w1,       // [E, D, H]
    const float* __restrict__ b1,       // [E, H]
    const float* __restrict__ w2,       // [E, H, O]
    const float* __restrict__ b2,       // [E, O]
    const int* __restrict__ counts,     // [E]
    const int* __restrict__ lists,      // [E, B]
    float* __restrict__ y_ws)           // [E*B, O]
{
    __shared__ float As[64 * ASTRIDE];  // gathered x tile
    __shared__ float Bs[KC * BSTRIDE];  // w1 chunk, H padded to 96 (zeros)
    __shared__ float Hs[64 * HSTRIDE];  // hidden activations (padded, zeros)
    __shared__ int   toks[64];
    float* W2s = As;                    // alias: 96*16 = 1536 floats <= 3648

    const int e    = blockIdx.x;
    const int tile = blockIdx.y;
    const int cnt  = counts[e];
    const int base = tile * 64;
    if (base >= cnt) return;            // block-uniform: whole wave exits

    const int tid  = threadIdx.x;
    const int lane = tid & 31;
    const int wave = tid >> 5;
    const int lrow = lane & 15;
    const int lhi  = lane >> 4;

    if (tid < 64) {
        int s = base + tid;
        toks[tid] = (s < cnt) ? lists[e * B_ + s] : 0;  // pad rows read token 0
    }
    __syncthreads();

    // ---- GEMM1: h[64,96] = relu(x_gather[64,784] @ w1[e] + b1[e]) ----
    v8f acc0 = {}, acc1 = {}, acc2 = {};   // 24 (mt,nt) pairs, 3 per wave

    for (int kc = 0; kc < NCHUNK; ++kc) {
        const int k0 = kc * KC;
        for (int i = tid; i < 64 * KC; i += 256) {
            int r = i / KC, c = i % KC;
            As[r * ASTRIDE + c] = x[(size_t)toks[r] * D_ + k0 + c];
        }
        for (int i = tid; i < KC * 96; i += 256) {
            int kk = i / 96, c = i % 96;
            Bs[kk * BSTRIDE + c] =
                (c < H_) ? w1[((size_t)e * D_ + (k0 + kk)) * H_ + c] : 0.0f;
        }
        __syncthreads();
#pragma unroll
        for (int p = 0; p < 3; ++p) {
            const int pair = wave * 3 + p;     // pair = nt*4 + mt
            const int mt = pair & 3, nt = pair >> 2;
            v8f* ap = (p == 0) ? &acc0 : (p == 1) ? &acc1 : &acc2;
            const float* ar = &As[(mt * 16 + lrow) * ASTRIDE + lhi * 2];
            const float* bc = &Bs[(lhi * 2) * BSTRIDE + nt * 16 + lrow];
#pragma unroll
            for (int ks = 0; ks < NCHUNK; ++ks) {
                v2f a, b;
                a.x = ar[ks * 4 + 0];
                a.y = ar[ks * 4 + 1];
                b.x = bc[(ks * 4 + 0) * BSTRIDE];
                b.y = bc[(ks * 4 + 1) * BSTRIDE];
                *ap = __builtin_amdgcn_wmma_f32_16x16x4_f32(
                        false, a, false, b, (short)0, *ap, false, false);
            }
        }
        __syncthreads();
    }

    // bias + relu -> Hs   (padded cols: acc==0, bias==0 -> stores 0)
#pragma unroll
    for (int p = 0; p < 3; ++p) {
        const int pair = wave * 3 + p;
        const int mt = pair & 3, nt = pair >> 2;
        v8f acc = (p == 0) ? acc0 : (p == 1) ? acc1 : acc2;
        const int col = nt * 16 + lrow;
        const float bias = (col < H_) ? b1[e * H_ + col] : 0.0f;
#pragma unroll
        for (int i = 0; i < 8; ++i) {
            int row = mt * 16 + i + lhi * 8;
            float v = acc[i] + bias;
            Hs[row * HSTRIDE + col] = v > 0.0f ? v : 0.0f;
        }
    }
    // stage w2 (zero-padded to [96][16]) into the region that aliased As
    for (int i = tid; i < 96 * 16; i += 256) {
        int kk = i >> 4, c = i & 15;
        W2s[i] = (kk < H_ && c < O_)
               ? w2[((size_t)e * H_ + kk) * O_ + c] : 0.0f;
    }
    __syncthreads();

    // ---- GEMM2: y[64,16] = Hs[64,96] @ W2s[96,16], waves 0..3 (mt = wave) ----
    if (wave < 4) {                       // wave-uniform branch: EXEC stays full
        v8f y = {};
        const float* hr = &Hs[(wave * 16 + lrow) * HSTRIDE + lhi * 2];
        const float* wc = &W2s[(lhi * 2) * 16 + lrow];
#pragma unroll
        for (int ks = 0; ks < 24; ++ks) {
            v2f a, b;
            a.x = hr[ks * 4 + 0];
            a.y = hr[ks * 4 + 1];
            b.x = wc[(ks * 4 + 0) * 16];
            b.y = wc[(ks * 4 + 1) * 16];
            y = __builtin_amdgcn_wmma_f32_16x16x4_f32(
                    false, a, false, b, (short)0, y, false, false);
        }
        if (lrow < O_) {
            const float bias = b2[e * O_ + lrow];
#pragma unroll
            for (int i = 0; i < 8; ++i) {
                int slot = base + wave * 16 + i + lhi * 8;
                if (slot < cnt)
                    y_ws[((size_t)e * B_ + slot) * O_ + lrow] = y[i] + bias;
            }
        }
    }
}

// ---------------------------------------------------------------------------
// combine: out[b] = w0 * y[pos0] + w1 * y[pos1]   (deterministic, no atomics)
// ---------------------------------------------------------------------------
__global__ void moe_combine_kernel(
    const float* __restrict__ y_ws,
    const int* __restrict__ pos,
    const float* __restrict__ topw,
    float* __restrict__ out)
{
    int b = blockIdx.x * blockDim.x + threadIdx.x;
    if (b >= B_) return;
    const int   p0 = pos[b * 2 + 0], p1 = pos[b * 2 + 1];
    const float w0 = topw[b * 2 + 0], w1 = topw[b * 2 + 1];
#pragma unroll
    for (int o = 0; o < O_; ++o) {
        out[(size_t)b * O_ + o] =
            w0 * y_ws[(size_t)p0 * O_ + o] + w1 * y_ws[(size_t)p1 * O_ + o];
    }
}

// ---------------------------------------------------------------------------
extern "C" void kernel_launch(void* const* d_in, const int* in_sizes, int n_in,
                              void* d_out, int out_size, void* d_ws, size_t ws_size,
                              hipStream_t stream) {
    const float* x  = (const float*)d_in[0];   // [B, D]
    const float* gw = (const float*)d_in[1];   // [D, E]
    const float* gb = (const float*)d_in[2];   // [E]
    const float* w1 = (const float*)d_in[3];   // [E, D, H]
    const float* b1 = (const float*)d_in[4];   // [E, H]
    const float* w2 = (const float*)d_in[5];   // [E, H, O]
    const float* b2 = (const float*)d_in[6];   // [E, O]

    float* out        = (float*)d_out;                 // [B*O]
    float* gate_probs = out + (size_t)B_ * O_;         // [B*E]

    // workspace layout (~35.2 MB)
    char* ws = (char*)d_ws;
    int*   counts = (int*)ws;                                       // 256 B
    int*   lists  = (int*)(ws + 256);                               // E*B int
    int*   pos    = (int*)(ws + 256 + (size_t)E_ * B_ * 4);         // B*2 int
    float* topw   = (float*)(ws + 256 + (size_t)E_ * B_ * 4
                                  + (size_t)B_ * 2 * 4);            // B*2 f32
    float* y_ws   = (float*)(ws + 256 + (size_t)E_ * B_ * 4
                                  + (size_t)B_ * 2 * 4
                                  + (size_t)B_ * 2 * 4);            // E*B*O f32

    moe_init_kernel<<<1, 32, 0, stream>>>(counts);
    moe_router_kernel<<<B_ / 64, 128, 0, stream>>>(
        x, gw, gb, gate_probs, counts, lists, pos, topw);
    moe_expert_kernel<<<dim3(E_, B_ / 64), 256, 0, stream>>>(
        x, w1, b1, w2, b2, counts, lists, y_ws);
    moe_combine_kernel<<<(B_ + 255) / 256, 256, 0, stream>>>(
        y_ws, pos, topw, out);
}